// gnn_nodes_15573551415676
// MI455X (gfx1250) — compile-verified
//
#include <hip/hip_runtime.h>
#include <hip/hip_bf16.h>

// B=4, N=512, F_IN=64, F_OUT=128
#define BB 4
#define NN 512
#define FIN 64
#define FOUT 128

typedef float v2f __attribute__((ext_vector_type(2)));
typedef float v4f __attribute__((ext_vector_type(4)));
typedef float v8f __attribute__((ext_vector_type(8)));

// ---------------------------------------------------------------------------
// Generic fp32 WMMA GEMM:  Y = act( X @ op(W) + bias ) [+ addM]
//   out[m, n] = sum_k A[m,k] * Bm[k,n]
//   A[m,k]  = TRANSA ? X[k*lda + m] : X[m*lda + k]
//   Bm[k,n] = TRANSW ? W[k*ldw + n] : W[n*ldw + k]      (W.T convention)
// One wave computes one 16x16 tile via V_WMMA_F32_16X16X4_F32.
// CDNA5 layouts (ISA 7.12.2):
//   A 16x4 : lanes 0-15 row M=l ; VGPR0 = K {0|2} per lane-half, VGPR1 = K {1|3}
//   B 4x16 : mirrored (rows striped across lanes, K {0|2}/{1|3} per half)
//   C/D    : VGPR r -> row m0 + r + 8*half, col n0 + (lane&15)
// All mode flags are template params -> branch-free, strength-reduced K-loop.
// ---------------------------------------------------------------------------
template <int TRANSA, int TRANSW, int ACT, bool HASBIAS, bool HASADD>
__global__ __launch_bounds__(32)
void wmma_gemm_f32(const float* __restrict__ X, long xbs, int lda,
                   const float* __restrict__ W, long wbs, int ldw,
                   const float* __restrict__ bias,
                   const float* __restrict__ addM,
                   float* __restrict__ Y, long ybs, int ldy,
                   int K)
{
    const int lane = threadIdx.x & 31;
    const int half = lane >> 4;     // 0: lanes 0-15, 1: lanes 16-31
    const int l    = lane & 15;
    const int m0   = blockIdx.x * 16;
    const int n0   = blockIdx.y * 16;
    const int b    = blockIdx.z;

    X += (long)b * xbs;
    W += (long)b * wbs;
    Y += (long)b * ybs;

    // Hoisted base pointers; K-loop uses pure pointer increments.
    const float* pA = TRANSA ? (X + (long)(2 * half) * lda + m0 + l)
                             : (X + (long)(m0 + l) * lda + 2 * half);
    const long   sA = TRANSA ? (long)4 * lda : 4;
    const float* pW = TRANSW ? (W + (long)(2 * half) * ldw + n0 + l)
                             : (W + (long)(n0 + l) * ldw + 2 * half);
    const long   sW = TRANSW ? (long)4 * ldw : 4;

    v8f acc = {};

#pragma unroll 4
    for (int k = 0; k < K; k += 4) {
        v2f a, bb;
        if (TRANSA) { a.x = pA[0];  a.y = pA[lda]; }
        else        { a  = *(const v2f*)pA; }
        if (TRANSW) { bb.x = pW[0]; bb.y = pW[ldw]; }
        else        { bb = *(const v2f*)pW; }
        pA += sA;
        pW += sW;
        // (neg_a, A, neg_b, B, c_mod, C, reuse_a, reuse_b)
        acc = __builtin_amdgcn_wmma_f32_16x16x4_f32(
                  false, a, false, bb, (short)0, acc, false, false);
    }

    const float bv = HASBIAS ? bias[n0 + l] : 0.0f;
#pragma unroll
    for (int r = 0; r < 8; ++r) {
        const int m = m0 + r + 8 * half;
        float v = acc[r] + bv;
        if (ACT == 1) v = (v > 0.0f) ? v : expm1f(v);   // CELU(alpha=1)
        if (HASADD) v += addM[(long)m * ldy + n0 + l];
        Y[(long)m * ldy + n0 + l] = v;
    }
}

// Scol[b*N + j] = sum_i sk[b,i,j]   (one thread per (b,j), coalesced over j)
__global__ __launch_bounds__(256)
void colsum_kernel(const float* __restrict__ sk, float* __restrict__ Scol)
{
    const int idx = blockIdx.x * 256 + threadIdx.x;       // 0..B*N-1
    const int b = idx >> 9;
    const int j = idx & (NN - 1);
    const float* p = sk + (long)b * NN * NN + j;
    float s = 0.0f;
    for (int i = 0; i < NN; ++i) s += p[(long)i * NN];
    Scol[idx] = s;
}

// u[b,j,f] = pj[b,j,f] * Scol[b,j] + T[b,j,f]
__global__ __launch_bounds__(256)
void fuse_u_kernel(const float* __restrict__ pj, const float* __restrict__ T,
                   const float* __restrict__ Scol, float* __restrict__ u)
{
    const long idx = (long)blockIdx.x * 256 + threadIdx.x;  // over B*N*FOUT
    const long row = idx >> 7;
    u[idx] = pj[idx] * Scol[row] + T[idx];
}

// edges[b,i,j,f] = (pj[b,j,f] + pib[b,i,f]) * sk[b,i,j]   (pib = pi + edge_b)
// One block per (b,i); 8 waves sweep j; one b128 NT store per (i,j) per wave.
__global__ __launch_bounds__(256)
void edges_stream_kernel(const float* __restrict__ pj, const float* __restrict__ pib,
                         const float* __restrict__ sk, float* __restrict__ edges)
{
    const int bi   = blockIdx.x;            // b*N + i
    const int b    = bi >> 9;
    const int wave = threadIdx.x >> 5;
    const int lane = threadIdx.x & 31;
    const int f4   = lane * 4;

    const v4f pv = *(const v4f*)(pib + (long)bi * FOUT + f4);   // loop-invariant
    const float* skrow = sk + (long)bi * NN;
    const float* pjrow = pj + ((long)b * NN) * FOUT + f4;
    float*       erow  = edges + (long)bi * NN * FOUT + f4;

#pragma unroll 2
    for (int j = wave; j < NN; j += 8) {
        const float s = skrow[j];                               // wave-uniform
        const v4f   q = *(const v4f*)(pjrow + (long)j * FOUT);
        v4f o;
        o.x = (q.x + pv.x) * s;
        o.y = (q.y + pv.y) * s;
        o.z = (q.z + pv.z) * s;
        o.w = (q.w + pv.w) * s;
        __builtin_nontemporal_store(o, (v4f*)(erow + (long)j * FOUT));
    }
}

extern "C" void kernel_launch(void* const* d_in, const int* in_sizes, int n_in,
                              void* d_out, int out_size, void* d_ws, size_t ws_size,
                              hipStream_t stream)
{
    const float* nodes  = (const float*)d_in[0];   // [4,512,64]
    const float* sk     = (const float*)d_in[1];   // [4,512,512,1]
    const float* edge_W = (const float*)d_in[2];   // [128,128]
    const float* edge_b = (const float*)d_in[3];   // [128]
    const float* node_W = (const float*)d_in[4];   // [128,128]
    const float* node_b = (const float*)d_in[5];   // [128]
    const float* mlp_W0 = (const float*)d_in[6];   // [128,64]
    const float* mlp_b0 = (const float*)d_in[7];
    const float* mlp_W1 = (const float*)d_in[8];   // [128,128]
    const float* mlp_b1 = (const float*)d_in[9];
    const float* mlp_W2 = (const float*)d_in[10];  // [128,128]
    const float* mlp_b2 = (const float*)d_in[11];

    const long ROWS = (long)BB * NN;               // 2048
    const long MATF = ROWS * FOUT;                 // 262144 floats

    float* ws   = (float*)d_ws;
    float* pj   = ws;              // [2048,128]
    float* pib  = ws + 1 * MATF;   // [2048,128] = pi + edge_b
    float* h0   = ws + 2 * MATF;
    float* h1   = ws + 3 * MATF;
    float* h    = ws + 4 * MATF;
    float* T    = ws + 5 * MATF;   // sk^T @ pib, batched
    float* u    = ws + 6 * MATF;
    float* Scol = ws + 7 * MATF;   // [2048]

    float* out1  = (float*)d_out;          // h + nodes_new : [2048,128]
    float* edges = (float*)d_out + MATF;   // [4,512,512,128]

    dim3 wblk(32, 1, 1);
    dim3 gFull(ROWS / 16, FOUT / 16, 1);   // (128, 8, 1)

    // 1) pj = nodes @ edge_W[:, :64].T
    wmma_gemm_f32<0,0,0,false,false><<<gFull, wblk, 0, stream>>>(
        nodes, 0, FIN, edge_W, 0, 2 * FIN, nullptr, nullptr, pj, 0, FOUT, FIN);
    // 2) pib = nodes @ edge_W[:, 64:].T + edge_b   (kofs folded into W pointer)
    wmma_gemm_f32<0,0,0,true,false><<<gFull, wblk, 0, stream>>>(
        nodes, 0, FIN, edge_W + FIN, 0, 2 * FIN, edge_b, nullptr, pib, 0, FOUT, FIN);
    // 3) h0 = celu(nodes @ mlp_W0.T + b0)
    wmma_gemm_f32<0,0,1,true,false><<<gFull, wblk, 0, stream>>>(
        nodes, 0, FIN, mlp_W0, 0, FIN, mlp_b0, nullptr, h0, 0, FOUT, FIN);
    // 4) h1 = celu(h0 @ mlp_W1.T + b1)
    wmma_gemm_f32<0,0,1,true,false><<<gFull, wblk, 0, stream>>>(
        h0, 0, FOUT, mlp_W1, 0, FOUT, mlp_b1, nullptr, h1, 0, FOUT, FOUT);
    // 5) h = h1 @ mlp_W2.T + b2
    wmma_gemm_f32<0,0,0,true,false><<<gFull, wblk, 0, stream>>>(
        h1, 0, FOUT, mlp_W2, 0, FOUT, mlp_b2, nullptr, h, 0, FOUT, FOUT);
    // 6) Scol[b,j] = sum_i sk[b,i,j]
    colsum_kernel<<<ROWS / 256, 256, 0, stream>>>(sk, Scol);
    // 7) T[b] = sk[b]^T @ pib[b]   (batched; transA on sk, direct-layout pib)
    dim3 gBat(NN / 16, FOUT / 16, BB);     // (32, 8, 4)
    wmma_gemm_f32<1,1,0,false,false><<<gBat, wblk, 0, stream>>>(
        sk, (long)NN * NN, NN, pib, (long)NN * FOUT, FOUT, nullptr, nullptr,
        T, (long)NN * FOUT, FOUT, NN);
    // 8) u = pj * Scol + T   (nodes_new pre-node_W; edge_b term folded via pib)
    fuse_u_kernel<<<MATF / 256, 256, 0, stream>>>(pj, T, Scol, u);
    // 9) out1 = u @ node_W.T + node_b + h
    wmma_gemm_f32<0,0,0,true,true><<<gFull, wblk, 0, stream>>>(
        u, 0, FOUT, node_W, 0, FOUT, node_b, h, out1, 0, FOUT, FOUT);
    // 10) stream the 512 MB edges tensor with NT b128 stores
    edges_stream_kernel<<<BB * NN, 256, 0, stream>>>(pj, pib, sk, edges);
}